// diff_DIR_75788992905247
// MI455X (gfx1250) — compile-verified
//
#include <hip/hip_runtime.h>
#include <hip/hip_bf16.h>

typedef __attribute__((ext_vector_type(2))) float v2f;
typedef __attribute__((ext_vector_type(8))) float v8f;

#define VOCAB_P1 100001
#define EMB_DIM  128
#define BATCH    256
#define NNZ_Q    2048
#define NNZ_D    262144
#define NT       7            // 7 tiles of 16 cols = 112 >= 100

// ---------------------------------------------------------------- zero fill
__global__ void zero4_kernel(float4* __restrict__ p, long long n4) {
    long long i = (long long)blockIdx.x * blockDim.x + threadIdx.x;
    long long stride = (long long)gridDim.x * blockDim.x;
    float4 z; z.x = 0.f; z.y = 0.f; z.z = 0.f; z.w = 0.f;
    for (; i < n4; i += stride) p[i] = z;
}

// ------------------------------------------------- MLP (WMMA) + scatter to d
// One wave handles 16 document nnz (an M=16 tile); 8 waves / block.
// GEMM1: A[16x128] (gathered emb rows) x W1[128x112pad] via V_WMMA_F32_16X16X4_F32.
// GEMM2: h[16x112] . W2 done in-register + cross-lane reduce.
__global__ __launch_bounds__(256) void tdv_scatter_kernel(
    const int*   __restrict__ d_rows,
    const int*   __restrict__ d_cols,
    const float* __restrict__ d_freqs,
    const float* __restrict__ emb,
    const float* __restrict__ W1,   // [128,100]
    const float* __restrict__ b1,   // [100]
    const float* __restrict__ W2,   // [100]
    const float* __restrict__ b2,   // [1]
    float*       __restrict__ dmat) // [VOCAB_P1, BATCH]
{
    // W1 packed into B-operand layout: Bl[(n*32+kb)*64 + 2*lane + j]
    //   = W1[4*kb + 2*(lane>>4) + j][n*16 + (lane&15)]  (zero pad col>=100)
    __shared__ float Bl[NT * 32 * 64];   // 57344 bytes

    const int tid = threadIdx.x;
    for (int i = tid; i < NT * 32 * 64; i += 256) {
        int n   = i / (32 * 64);
        int kb  = (i >> 6) & 31;
        int L   = (i & 63) >> 1;
        int j   = i & 1;
        int col = n * 16 + (L & 15);
        int k   = kb * 4 + (L >> 4) * 2 + j;
        Bl[i] = (col < 100) ? W1[k * 100 + col] : 0.0f;
    }
    __syncthreads();

    const int wave = tid >> 5;
    const int lane = tid & 31;
    const int m    = lane & 15;      // matrix row this lane feeds (A operand)
    const int kh   = lane >> 4;      // K-half select
    const int g0   = (blockIdx.x * 8 + wave) * 16;   // first nnz of this tile

    const int row = d_rows[g0 + m];
    const float* erow = emb + (long long)row * EMB_DIM + 2 * kh;

    v8f c[NT] = {};

#pragma unroll 2
    for (int kb = 0; kb < 32; ++kb) {
        v2f a = *(const v2f*)(erow + kb * 4);   // A[m][4kb+2kh .. +1]
#pragma unroll
        for (int n = 0; n < NT; ++n) {
            v2f b = *(const v2f*)&Bl[(n * 32 + kb) * 64 + 2 * lane];
            c[n] = __builtin_amdgcn_wmma_f32_16x16x4_f32(
                false, a, false, b, (short)0, c[n], false, false);
        }
    }

    // GEMM2: relu(h + b1) . W2 ; C/D layout: lane half kh holds rows i+8*kh in c[n][i],
    // column = n*16 + (lane&15).
    const int col0 = lane & 15;
    float part[8];
#pragma unroll
    for (int i = 0; i < 8; ++i) part[i] = 0.0f;
#pragma unroll
    for (int n = 0; n < NT; ++n) {
        int col  = n * 16 + col0;
        bool ok  = (col < 100);
        float bb = ok ? b1[col] : 0.0f;
        float w2 = ok ? W2[col] : 0.0f;
#pragma unroll
        for (int i = 0; i < 8; ++i) {
            float h = c[n][i] + bb;
            h = (h > 0.0f) ? h : 0.0f;
            part[i] += h * w2;                 // w2==0 kills padded cols
        }
    }
    // reduce over the 16 lanes of each half (masks < 16 stay inside the half)
#pragma unroll
    for (int i = 0; i < 8; ++i) {
        float s = part[i];
        s += __shfl_xor(s, 1, 32);
        s += __shfl_xor(s, 2, 32);
        s += __shfl_xor(s, 4, 32);
        s += __shfl_xor(s, 8, 32);
        part[i] = s;
    }
    const float bias2 = b2[0];
    if (col0 == 0) {              // lanes 0 (rows 0-7) and 16 (rows 8-15)
#pragma unroll
        for (int i = 0; i < 8; ++i) {
            int   gg  = g0 + i + 8 * kh;
            float tdv = part[i] + bias2;
            tdv = (tdv > 0.0f) ? tdv : 0.0f;
            float val = tdv * d_freqs[gg];
            atomicAdd(&dmat[(long long)d_rows[gg] * BATCH + d_cols[gg]], val);
        }
    }
}

// --------------------------------------------------------------- column sums
__global__ __launch_bounds__(256) void colsum_kernel(
    const float* __restrict__ dmat, float* __restrict__ colsum)
{
    const int b = threadIdx.x;          // column index, coalesced reads
    float acc = 0.0f;
    for (int v = blockIdx.x; v < VOCAB_P1; v += gridDim.x)
        acc += dmat[(long long)v * BATCH + b];
    atomicAdd(&colsum[b], acc);
}

// -------------------------------------------------- row sums for query rows
__global__ __launch_bounds__(256) void rowsumq_kernel(
    const int* __restrict__ q_rows, const float* __restrict__ dmat,
    float* __restrict__ rowsumq)
{
    const int wid  = ((blockIdx.x * blockDim.x + threadIdx.x) >> 5);
    const int lane = threadIdx.x & 31;
    const int nw   = (gridDim.x * blockDim.x) >> 5;
    for (int i = wid; i < NNZ_Q; i += nw) {
        const float* r = dmat + (long long)q_rows[i] * BATCH;
        float acc = 0.0f;
#pragma unroll
        for (int b = lane; b < BATCH; b += 32) acc += r[b];
        acc += __shfl_xor(acc, 1, 32);
        acc += __shfl_xor(acc, 2, 32);
        acc += __shfl_xor(acc, 4, 32);
        acc += __shfl_xor(acc, 8, 32);
        acc += __shfl_xor(acc, 16, 32);
        if (lane == 0) rowsumq[i] = acc;
    }
}

// -------------------------------------------------------------- finalize rel
__global__ __launch_bounds__(256) void final_kernel(
    const int* __restrict__ q_rows, const int* __restrict__ q_cols,
    const float* __restrict__ q_vals, const float* __restrict__ dmat,
    const float* __restrict__ colsum, const float* __restrict__ rowsumq,
    const float* __restrict__ mu_p, float* __restrict__ rel)
{
    __shared__ float red[BATCH];
    __shared__ float smooth[BATCH];
    const int t = threadIdx.x;
    const float mu = mu_p[0];

    red[t] = colsum[t];
    __syncthreads();
    for (int s = 128; s > 0; s >>= 1) {
        if (t < s) red[t] += red[t + s];
        __syncthreads();
    }
    const float total = red[0];
    smooth[t] = logf(mu / (colsum[t] + mu));
    __syncthreads();

    for (int i = t; i < NNZ_Q; i += 256) {
        int   r  = q_rows[i];
        int   cc = q_cols[i];
        float dv = dmat[(long long)r * BATCH + cc];
        float cf = rowsumq[i] / total;
        float dir = log1pf(dv / (1.0f + mu * cf)) + smooth[cc];
        atomicAdd(&rel[cc], q_vals[i] * dir);
    }
}

// ---------------------------------------------------------------------------
extern "C" void kernel_launch(void* const* d_in, const int* in_sizes, int n_in,
                              void* d_out, int out_size, void* d_ws, size_t ws_size,
                              hipStream_t stream) {
    const int*   q_rows  = (const int*)  d_in[0];
    const int*   q_cols  = (const int*)  d_in[1];
    const float* q_vals  = (const float*)d_in[2];
    const int*   d_rows  = (const int*)  d_in[3];
    const int*   d_cols  = (const int*)  d_in[4];
    const float* d_freqs = (const float*)d_in[5];
    const float* emb     = (const float*)d_in[6];
    const float* W1      = (const float*)d_in[7];
    const float* b1      = (const float*)d_in[8];
    const float* W2      = (const float*)d_in[9];
    const float* b2      = (const float*)d_in[10];
    const float* mu      = (const float*)d_in[11];
    (void)in_sizes; (void)n_in; (void)out_size; (void)ws_size;

    float* rel  = (float*)d_out;          // [256]
    float* dmat = (float*)d_out + BATCH;  // [VOCAB_P1, BATCH]

    float* colsum  = (float*)d_ws;        // [256]
    float* rowsumq = colsum + BATCH;      // [2048]

    // zero outputs + colsum accumulator
    const long long out_n4 = (long long)(BATCH + (long long)VOCAB_P1 * BATCH) / 4;
    zero4_kernel<<<4096, 256, 0, stream>>>((float4*)d_out, out_n4);
    zero4_kernel<<<1, 128, 0, stream>>>((float4*)d_ws, (BATCH + NNZ_Q) / 4);

    // MLP + scatter: 262144 nnz / (8 waves * 16) = 2048 blocks
    tdv_scatter_kernel<<<2048, 256, 0, stream>>>(
        d_rows, d_cols, d_freqs, emb, W1, b1, W2, b2, dmat);

    colsum_kernel<<<512, 256, 0, stream>>>(dmat, colsum);
    rowsumq_kernel<<<64, 256, 0, stream>>>(q_rows, dmat, rowsumq);

    final_kernel<<<1, 256, 0, stream>>>(q_rows, q_cols, q_vals, dmat,
                                        colsum, rowsumq, mu, rel);
}